// LIF_Activation_64673617543596
// MI455X (gfx1250) — compile-verified
//
#include <hip/hip_runtime.h>
#include <cstddef>

// LIF activation over [B=128, T=1000, C=512] float32.
// Memory-bound streaming recurrence: 524 MB moved once -> ~22.5 us floor at
// 23.3 TB/s. No matrix structure, so no WMMA; the optimization target is a
// full-rate HBM stream with deep per-wave load pipelining (wave32, LOADcnt
// depth) and non-temporal cache policy (data has zero reuse, > L2 capacity).

namespace {

constexpr int kB = 128;
constexpr int kT = 1000;
constexpr int kC = 512;
constexpr int kU = 25;              // timesteps per pipeline stage (regs)
constexpr int kChunks = kT / kU;    // 40 (even -> clean 2x-unrolled pipeline)

__device__ __forceinline__ void load_chunk(float (&buf)[kU], const float* __restrict__ p) {
#pragma unroll
  for (int u = 0; u < kU; ++u) {
    // Stride between timesteps is C*4 = 2048 B -> fits the 24-bit immediate
    // offset, so this is kU back-to-back global_load_b32 ... th:TH_LOAD_NT
    // off one base pointer, all in flight before the first consumer waits.
    buf[u] = __builtin_nontemporal_load(p + (size_t)u * kC);
  }
}

__device__ __forceinline__ void compute_chunk(float (&buf)[kU], float& Vm,
                                              float decay, float* __restrict__ o) {
#pragma unroll
  for (int u = 0; u < kU; ++u) {
    // gate = (1 - Vm > 0) == (Vm < 1) exactly for Vm >= 0 (post-relu).
    // Fold the 0/1 gate into the decay factor: multiplying by exactly 1.0 or
    // 0.0 is exact, so g*Vm matches ref's (decay*Vm)*gate bit-for-bit.
    const float g = (Vm < 1.0f) ? decay : 0.0f;
    // Keep mul and add as two rounded ops (no FMA contraction) to match the
    // reference's rounding; spike thresholds sit on exact compares.
    Vm = fmaxf(0.0f, __fadd_rn(buf[u], __fmul_rn(g, Vm)));
    const float s = (Vm > 1.0f) ? 1.0f : 0.0f;   // (Vm - 1 > 0) == (Vm > 1)
    __builtin_nontemporal_store(s, o + (size_t)u * kC);
  }
}

__global__ __launch_bounds__(256) void lif_scan_kernel(const float* __restrict__ x,
                                                       const float* __restrict__ w_leak,
                                                       float* __restrict__ out) {
  const int idx = blockIdx.x * 256 + threadIdx.x;  // 0 .. B*C-1, one (b,c) column
  const int c = idx & (kC - 1);
  const int b = idx >> 9;                          // idx / 512
  const float decay = 1.0f - w_leak[c];

  const size_t base = (size_t)b * kT * kC + c;     // element (b, t=0, c)
  const float* xp = x + base;
  float* op = out + base;

  float buf0[kU], buf1[kU];
  float Vm = 0.0f;

  // Software pipeline: while computing one chunk of 25 timesteps, the next
  // chunk's 25 loads are already outstanding. Outer loop unrolled by 2 so the
  // double buffers are statically addressed (stay in VGPRs, no scratch).
  load_chunk(buf0, xp);
  for (int k = 0; k < kChunks; k += 2) {
    const float* xk = xp + (size_t)k * kU * kC;
    float* ok = op + (size_t)k * kU * kC;

    load_chunk(buf1, xk + (size_t)kU * kC);        // prefetch chunk k+1
    compute_chunk(buf0, Vm, decay, ok);            // consume chunk k

    if (k + 2 < kChunks) {
      load_chunk(buf0, xk + (size_t)(2 * kU) * kC);  // prefetch chunk k+2
    }
    compute_chunk(buf1, Vm, decay, ok + (size_t)kU * kC);  // consume chunk k+1
  }
}

}  // namespace

extern "C" void kernel_launch(void* const* d_in, const int* in_sizes, int n_in,
                              void* d_out, int out_size, void* d_ws, size_t ws_size,
                              hipStream_t stream) {
  (void)in_sizes; (void)n_in; (void)out_size; (void)d_ws; (void)ws_size;
  const float* x = (const float*)d_in[0];       // [B, T, C] float32
  const float* w_leak = (const float*)d_in[1];  // [C] float32
  float* out = (float*)d_out;                   // [B, T, C] float32 (spikes)

  const int threads = kB * kC;                  // one thread per (b, c)
  lif_scan_kernel<<<threads / 256, 256, 0, stream>>>(x, w_leak, out);
}